// Skeleton_61976378082048
// MI455X (gfx1250) — compile-verified
//
#include <hip/hip_runtime.h>

// FK-as-GEMM for gfx1250:  out[B,72] = ori[B,216] @ M[216,72]
// M is built on-device from offsets (tree chains + both G-permutations folded in).
// Bandwidth-bound (603 MB total @ 23.3 TB/s ~ 26us floor); WMMA f32 16x16x4 makes
// the 16.3 GFLOP of dense math free.

typedef __attribute__((ext_vector_type(2))) float v2f;
typedef __attribute__((ext_vector_type(8))) float v8f;

#define NJ 24
#define KDIM 216      // 24*9 input floats per body
#define NDIM 72       // 24*3 output floats per body
#define MT_STRIDE 218 // padded K stride for LDS bank spread (8B aligned)
#define MT_ELEMS (NDIM * MT_STRIDE)

__constant__ int c_parents[NJ] = {-1, 0, 0, 0, 1, 2, 3, 4, 5, 6, 7, 8,
                                  9, 9, 9, 12, 13, 14, 16, 17, 18, 19, 20, 21};

// Build M^T[72][218] into global scratch. Column n = (joint i, axis o):
//   out[b,i,o] = pos[b,i,g(o)],  pos[i] += R[b,p,q(r'),q(c')] * offsets[i,c'] along chain
// with q = (2,0,1) (from G^T R G) and g = (1,2,0) (from final G @ pos).
__global__ void build_mt_kernel(const float* __restrict__ offsets,
                                float* __restrict__ mt) {
    const int n = threadIdx.x;           // 0..71
    if (n >= NDIM) return;
    float* row = mt + (size_t)n * MT_STRIDE;
    for (int k = 0; k < MT_STRIDE; ++k) row[k] = 0.0f;

    const int q[3] = {2, 0, 1};
    const int g[3] = {1, 2, 0};
    const int joint = n / 3;
    const int o = n % 3;
    const int rp = g[o];                 // which pos component feeds this output
    int a = joint;
    while (a != 0) {                     // walk chain up to root
        const int p = c_parents[a];
        #pragma unroll
        for (int c = 0; c < 3; ++c) {
            const int k = p * 9 + q[rp] * 3 + q[c];
            row[k] += offsets[a * 3 + c];
        }
        a = p;
    }
}

__global__ void __launch_bounds__(256)
fk_wmma_kernel(const float* __restrict__ A,   // [B,216]
               const float* __restrict__ mt,  // M^T [72][218]
               float* __restrict__ C,         // [B,72]
               int nbatch) {
    __shared__ float sMt[MT_ELEMS];            // 62,784 bytes
    for (int i = threadIdx.x; i < MT_ELEMS; i += 256) sMt[i] = mt[i];
    __syncthreads();

    const int lane = threadIdx.x & 31;
    const int wave = threadIdx.x >> 5;
    const int row  = lane & 15;                // M index (A) / N index (B,C)
    const int hi   = lane >> 4;                // K half-select per ISA layout
    const int tile = blockIdx.x * 8 + wave;    // 16 batch rows per wave
    if (tile * 16 >= nbatch) return;           // wave-uniform guard (EXEC stays all-1)

    // A-frag source: lane holds A[tile*16+row][k0 + 2*hi + {0,1}] as float2
    const float* arow = A + (size_t)(tile * 16 + row) * KDIM + hi * 2;

    // B-frag sources: lane holds M[k0 + 2*hi + {0,1}][n0 + row] == sMt[n0+row][k..k+1]
    const float* b0p = &sMt[(0 * 16 + row) * MT_STRIDE + hi * 2];
    const float* b1p = &sMt[(1 * 16 + row) * MT_STRIDE + hi * 2];
    const float* b2p = &sMt[(2 * 16 + row) * MT_STRIDE + hi * 2];
    const float* b3p = &sMt[(3 * 16 + row) * MT_STRIDE + hi * 2];
    // N-tile 4 covers cols 64..79; alias 72..79 -> 64..71 (results never stored)
    const float* b4p = &sMt[(64 + (row & 7)) * MT_STRIDE + hi * 2];

    v8f acc0 = {}, acc1 = {}, acc2 = {}, acc3 = {}, acc4 = {};

    #pragma unroll 6
    for (int k0 = 0; k0 < KDIM; k0 += 4) {
        const v2f a  = *(const v2f*)(arow + k0);
        const v2f b0 = *(const v2f*)(b0p + k0);
        const v2f b1 = *(const v2f*)(b1p + k0);
        const v2f b2 = *(const v2f*)(b2p + k0);
        const v2f b3 = *(const v2f*)(b3p + k0);
        const v2f b4 = *(const v2f*)(b4p + k0);
        acc0 = __builtin_amdgcn_wmma_f32_16x16x4_f32(false, a, false, b0, (short)0, acc0, false, false);
        acc1 = __builtin_amdgcn_wmma_f32_16x16x4_f32(false, a, false, b1, (short)0, acc1, false, false);
        acc2 = __builtin_amdgcn_wmma_f32_16x16x4_f32(false, a, false, b2, (short)0, acc2, false, false);
        acc3 = __builtin_amdgcn_wmma_f32_16x16x4_f32(false, a, false, b3, (short)0, acc3, false, false);
        acc4 = __builtin_amdgcn_wmma_f32_16x16x4_f32(false, a, false, b4, (short)0, acc4, false, false);
    }

    // C layout: VGPR v -> row (v + 8*hi), col = lane&15. 16-lane 64B contiguous stores.
    #pragma unroll
    for (int v = 0; v < 8; ++v) {
        float* crow = C + (size_t)(tile * 16 + v + 8 * hi) * NDIM;
        crow[0  + row] = acc0[v];
        crow[16 + row] = acc1[v];
        crow[32 + row] = acc2[v];
        crow[48 + row] = acc3[v];
        if (row < 8) crow[64 + row] = acc4[v];   // only cols 64..71 are real
    }
}

extern "C" void kernel_launch(void* const* d_in, const int* in_sizes, int n_in,
                              void* d_out, int out_size, void* d_ws, size_t ws_size,
                              hipStream_t stream) {
    const float* orientations = (const float*)d_in[0];  // [B,24,3,3] f32
    const float* offsets      = (const float*)d_in[1];  // [24,3] f32
    float* out = (float*)d_out;                          // [B,24,3] f32
    float* mt  = (float*)d_ws;                           // M^T scratch (62.8 KB)

    const int nbatch = in_sizes[0] / (NJ * 9);

    build_mt_kernel<<<1, NDIM, 0, stream>>>(offsets, mt);

    const int tiles  = (nbatch + 15) / 16;   // 16 batch rows per wave
    const int blocks = (tiles + 7) / 8;      // 8 waves per block
    fk_wmma_kernel<<<blocks, 256, 0, stream>>>(orientations, mt, out, nbatch);
}